// MTRLiteLoss_40029095198849
// MI455X (gfx1250) — compile-verified
//
#include <hip/hip_runtime.h>
#include <hip/hip_bf16.h>
#include <stdint.h>

// Problem constants (match reference)
#define L_ 6
#define B_ 32
#define T_ 64
#define K_ 64
#define F_ 80
#define BT_ (B_*T_)
#define BLOCKS_PER_LAYER 32
#define NBLK (L_*BLOCKS_PER_LAYER)

// Workspace layout (32-bit word indices)
#define OFF_NVALID 0
#define OFF_VALID  16
#define OFF_INVN   (OFF_VALID + BT_)
#define OFF_GT     (OFF_INVN + BT_)
#define OFF_PART   (OFF_GT + 2*BT_)        // NBLK * 2 floats of block partials
#define OFF_VBITS  (OFF_PART + 2*NBLK)     // 3 packed mask words per (b,t)

#ifndef __has_builtin
#define __has_builtin(x) 0
#endif

#if __has_builtin(__builtin_amdgcn_global_load_async_to_lds_b128)
#define USE_ASYNC_BUILTIN 1
#else
#define USE_ASYNC_BUILTIN 0
#endif

#if __has_builtin(__builtin_amdgcn_s_wait_asynccnt)
#define USE_WAIT_BUILTIN 1
#else
#define USE_WAIT_BUILTIN 0
#endif

typedef int v4i __attribute__((ext_vector_type(4)));
typedef __attribute__((address_space(1))) v4i g_v4i;   // global (AS1) int4
typedef __attribute__((address_space(3))) v4i l_v4i;   // LDS (AS3) int4

// Async copy 16B/lane from global to this wave's LDS slice (CDNA5 async path, ASYNCcnt)
__device__ __forceinline__ void stage16(const float* g, float* lds_generic) {
#if USE_ASYNC_BUILTIN
  __builtin_amdgcn_global_load_async_to_lds_b128((g_v4i*)g, (l_v4i*)lds_generic, 0, 0);
#else
  unsigned lds_off = (unsigned)(size_t)lds_generic;   // low 32 bits of flat LDS addr = LDS offset
  asm volatile("global_load_async_to_lds_b128 %0, %1, off"
               :: "v"(lds_off), "v"(g) : "memory");
#endif
}

__device__ __forceinline__ void wait_async0() {
#if USE_WAIT_BUILTIN
  __builtin_amdgcn_s_wait_asynccnt(0);
#else
  asm volatile("s_wait_asynccnt 0x0" ::: "memory");
#endif
  asm volatile("" ::: "memory");  // compiler ordering fence before LDS reads
}

__device__ __forceinline__ float wave_sum(float v) {
#pragma unroll
  for (int o = 16; o > 0; o >>= 1) v += __shfl_xor(v, o, 32);
  return v;
}

__device__ __forceinline__ float wave_max(float v) {
#pragma unroll
  for (int o = 16; o > 0; o >>= 1) v = fmaxf(v, __shfl_xor(v, o, 32));
  return v;
}

__device__ __forceinline__ float sl1(float x) {
  float a = fabsf(x);
  return (a < 1.0f) ? 0.5f * x * x : a - 0.5f;
}

// ---------------- kernel 1: per-(l,b,t) item, one wave each ------------------
__global__ __launch_bounds__(256) void k_main(const float* __restrict__ trajs,
                                              const float* __restrict__ scores,
                                              const float* __restrict__ tfut,
                                              float* __restrict__ ws) {
  __shared__ __align__(16) float stage[8][336];  // per wave: [0..159]=best row, [160..319]=target row
  __shared__ float wacc[8][2];

  const int layer = blockIdx.x / BLOCKS_PER_LAYER;
  const int blkl  = blockIdx.x % BLOCKS_PER_LAYER;
  const int wave  = threadIdx.x >> 5;
  const int lane  = threadIdx.x & 31;
  const int wave_global = blkl * 8 + wave;  // 0..255 within layer
  const unsigned* uws = (const unsigned*)ws;

  float cls_acc = 0.0f, reg_acc = 0.0f;

  float* lbest = &stage[wave][0];
  float* ltgt  = &stage[wave][160];

#pragma unroll 1
  for (int i = 0; i < 8; ++i) {
    const int it = wave_global + 256 * i;              // (b,t) flat, 0..2047
    const size_t lbt = (size_t)layer * BT_ + it;       // (l,b,t) flat
    const float* trj = trajs + lbt * (size_t)(K_ * F_ * 2);
    const float* sc  = scores + lbt * K_;

    // Prefetch next item's scores + endpoint lines (global_prefetch_b8, speculative)
    if (it + 256 < BT_) {
      const size_t lbt2 = lbt + 256;
      const float* sc2  = scores + lbt2 * K_;
      const float* trj2 = trajs + lbt2 * (size_t)(K_ * F_ * 2);
      __builtin_prefetch(sc2 + lane, 0, 3);
      __builtin_prefetch(trj2 + ((size_t)lane * F_ + (F_ - 1)) * 2, 0, 3);
      __builtin_prefetch(trj2 + ((size_t)(lane + 32) * F_ + (F_ - 1)) * 2, 0, 3);
    }

    const float gx = ws[OFF_GT + 2 * it];
    const float gy = ws[OFF_GT + 2 * it + 1];

    // ---- endpoint distances + argmin over K=64 (lane handles k=lane, lane+32)
    const float* ep0 = trj + ((size_t)lane * F_ + (F_ - 1)) * 2;
    const float* ep1 = trj + ((size_t)(lane + 32) * F_ + (F_ - 1)) * 2;
    float dx0 = ep0[0] - gx, dy0 = ep0[1] - gy;
    float dx1 = ep1[0] - gx, dy1 = ep1[1] - gy;
    float d0 = dx0 * dx0 + dy0 * dy0;
    float d1 = dx1 * dx1 + dy1 * dy1;
    float dmin = (d0 <= d1) ? d0 : d1;            // tie -> lower k (matches argmin)
    int   kmin = (d0 <= d1) ? lane : lane + 32;
#pragma unroll
    for (int o = 16; o > 0; o >>= 1) {
      float od = __shfl_xor(dmin, o, 32);
      int   ok = __shfl_xor(kmin, o, 32);
      if (od < dmin || (od == dmin && ok < kmin)) { dmin = od; kmin = ok; }
    }

    // ---- kick off async global->LDS staging of best row + target row (640B each)
    const float* row = trj + (size_t)kmin * (F_ * 2);
    const float* tg  = tfut + (size_t)it * (F_ * 2);
    stage16(row + lane * 4, lbest + lane * 4);            // 128 floats
    stage16(tg  + lane * 4, ltgt  + lane * 4);
    if (lane < 8) {                                       // remaining 32 floats each
      stage16(row + 128 + lane * 4, lbest + 128 + lane * 4);
      stage16(tg  + 128 + lane * 4, ltgt  + 128 + lane * 4);
    }

    // ---- logsumexp over scores (hides async copy latency)
    float s0 = sc[lane], s1 = sc[lane + 32];
    float m = wave_max(fmaxf(s0, s1));
    float e = expf(s0 - m) + expf(s1 - m);
    float logz = m + logf(wave_sum(e));
    float cand = (kmin & 32) ? s1 : s0;
    float best_score = __shfl(cand, kmin & 31, 32);
    float ce = logz - best_score;

    // ---- packed per-f valid masks (3 words, precomputed in k_prep)
    unsigned w0 = uws[OFF_VBITS + 3 * it];
    unsigned w1 = uws[OFF_VBITS + 3 * it + 1];
    unsigned w2 = uws[OFF_VBITS + 3 * it + 2];

    // ---- wait for staged rows, masked smooth-L1 from LDS (f = lane, lane+32, lane+64)
    wait_async0();
    float acc = 0.0f;
    {
      int f = lane;
      float ddx = lbest[2 * f] - ltgt[2 * f];
      float ddy = lbest[2 * f + 1] - ltgt[2 * f + 1];
      acc += ((w0 >> lane) & 1u) ? (sl1(ddx) + sl1(ddy)) : 0.0f;
    }
    {
      int f = lane + 32;
      float ddx = lbest[2 * f] - ltgt[2 * f];
      float ddy = lbest[2 * f + 1] - ltgt[2 * f + 1];
      acc += ((w1 >> lane) & 1u) ? (sl1(ddx) + sl1(ddy)) : 0.0f;
    }
    if (lane < 16) {
      int f = lane + 64;
      float ddx = lbest[2 * f] - ltgt[2 * f];
      float ddy = lbest[2 * f + 1] - ltgt[2 * f + 1];
      acc += ((w2 >> lane) & 1u) ? (sl1(ddx) + sl1(ddy)) : 0.0f;
    }
    float reg = wave_sum(acc) * ws[OFF_INVN + it];

    float vmask = ws[OFF_VALID + it];
    cls_acc += ce * vmask;
    reg_acc += reg * vmask;
  }

  // ---- deterministic block reduction of the 8 wave partials
  if (lane == 0) { wacc[wave][0] = cls_acc; wacc[wave][1] = reg_acc; }
  __syncthreads();
  if (threadIdx.x == 0) {
    float c = 0.0f, r = 0.0f;
    for (int w = 0; w < 8; ++w) { c += wacc[w][0]; r += wacc[w][1]; }
    ws[OFF_PART + blockIdx.x * 2]     = c;
    ws[OFF_PART + blockIdx.x * 2 + 1] = r;
  }
}

// ---------------- kernel 0: per-(b,t) precompute + n_valid -------------------
__global__ __launch_bounds__(256) void k_prep(const float* __restrict__ tfut,
                                              const int*   __restrict__ tfv,
                                              const int*   __restrict__ tmask,
                                              float* __restrict__ ws) {
  __shared__ float cnt[256];
  unsigned* uws = (unsigned*)ws;
  float local = 0.0f;
  for (int it = threadIdx.x; it < BT_; it += 256) {
    const int* vf = tfv + (size_t)it * F_;
    int s = 0;
    unsigned mw[3] = {0u, 0u, 0u};
#pragma unroll 8
    for (int f = 0; f < F_; ++f) {
      unsigned v = (vf[f] != 0) ? 1u : 0u;
      s += (int)v;
      mw[f >> 5] |= v << (f & 31);
    }
    uws[OFF_VBITS + 3 * it]     = mw[0];
    uws[OFF_VBITS + 3 * it + 1] = mw[1];
    uws[OFF_VBITS + 3 * it + 2] = mw[2];
    int valid = (tmask[it] != 0) && (s > 0);
    ws[OFF_VALID + it] = valid ? 1.0f : 0.0f;
    ws[OFF_INVN + it]  = 1.0f / fmaxf(2.0f * (float)s, 1.0f);
    const float* ge = tfut + ((size_t)it * F_ + (F_ - 1)) * 2;
    ws[OFF_GT + 2 * it]     = ge[0];
    ws[OFF_GT + 2 * it + 1] = ge[1];
    local += (float)valid;
  }
  cnt[threadIdx.x] = local;
  __syncthreads();
  for (int s = 128; s > 0; s >>= 1) {
    if ((int)threadIdx.x < s) cnt[threadIdx.x] += cnt[threadIdx.x + s];
    __syncthreads();
  }
  if (threadIdx.x == 0) ws[OFF_NVALID] = cnt[0];
}

// ---------------- kernel 2: final fixed-order combine ------------------------
__global__ void k_final(const float* __restrict__ ws, float* __restrict__ out) {
  if (threadIdx.x == 0 && blockIdx.x == 0) {
    const float wl[L_] = {1.0f/9.0f, 1.0f/9.0f, 1.0f/9.0f, 2.0f/9.0f, 2.0f/9.0f, 2.0f/9.0f};
    float nv = fmaxf(ws[OFF_NVALID], 1.0f);
    float tot = 0.0f;
    for (int l = 0; l < L_; ++l) {
      float c = 0.0f, r = 0.0f;
      for (int bb = 0; bb < BLOCKS_PER_LAYER; ++bb) {
        c += ws[OFF_PART + (l * BLOCKS_PER_LAYER + bb) * 2];
        r += ws[OFF_PART + (l * BLOCKS_PER_LAYER + bb) * 2 + 1];
      }
      float denom = (l == L_ - 1) ? nv : 1.0f;
      tot += wl[l] * ((c + r) / denom);   // CLS_WEIGHT = REG_WEIGHT = 1
    }
    out[0] = tot;
  }
}

extern "C" void kernel_launch(void* const* d_in, const int* in_sizes, int n_in,
                              void* d_out, int out_size, void* d_ws, size_t ws_size,
                              hipStream_t stream) {
  (void)in_sizes; (void)n_in; (void)out_size; (void)ws_size;
  const float* trajs  = (const float*)d_in[0];
  const float* scores = (const float*)d_in[1];
  const float* tfut   = (const float*)d_in[2];
  const int*   tfv    = (const int*)d_in[3];
  const int*   tmask  = (const int*)d_in[4];
  float* ws  = (float*)d_ws;
  float* out = (float*)d_out;

  k_prep<<<1, 256, 0, stream>>>(tfut, tfv, tmask, ws);
  k_main<<<NBLK, 256, 0, stream>>>(trajs, scores, tfut, ws);
  k_final<<<1, 32, 0, stream>>>(ws, out);
}